// PTransE_20873541059102
// MI455X (gfx1250) — compile-verified
//
#include <hip/hip_runtime.h>

#define BATCH   16384
#define NUM_REL 1200
#define DIM     128
#define KCHUNK  300              // 1200 = 4 * 300 ; LDS chunk = 300*128*4 = 153600 B

typedef __attribute__((ext_vector_type(2))) float v2f;
typedef __attribute__((ext_vector_type(8))) float v8f;

// ---------------------------------------------------------------------------
// Kernel 1: zero the relation_weight scratch (16384 x 1200 f32 in d_ws)
// ---------------------------------------------------------------------------
__global__ __launch_bounds__(256) void ptranse_zero(float4* __restrict__ p, int n4) {
    int i = blockIdx.x * blockDim.x + threadIdx.x;
    int stride = gridDim.x * blockDim.x;
    float4 z = make_float4(0.f, 0.f, 0.f, 0.f);
    for (; i < n4; i += stride) p[i] = z;
}

// ---------------------------------------------------------------------------
// Kernel 2: ragged weighted histogram via global f32 atomics (L2-resident)
// ---------------------------------------------------------------------------
__global__ __launch_bounds__(256) void ptranse_scatter(const int* __restrict__ pb,
                                                       const int* __restrict__ pr,
                                                       const float* __restrict__ prob,
                                                       float* __restrict__ rw, int n) {
    int i = blockIdx.x * blockDim.x + threadIdx.x;
    if (i < n) {
        atomicAdd(rw + (size_t)pb[i] * NUM_REL + (size_t)pr[i], prob[i]);
    }
}

// ---------------------------------------------------------------------------
// Kernel 3: P = rw @ rel_emb via V_WMMA_F32_16X16X4_F32, fused |P - rel_emb[r]|
//
// Block = 256 threads = 8 wave32; wave w owns M-tile (blockIdx.x*8 + w)
// (16 rows) across all 8 N-tiles (DIM = 128).
//
// B (rel_emb) staged through LDS in 4 chunks of K=300 (153.6 KB dynamic LDS)
// in a K-pair-interleaved layout:
//     bs[(p*DIM + n)*2 + i] = rel_emb[kbase + 2p + i][n],  i in {0,1}
// so each lane's B fragment (two consecutive-K values at one column) is a
// single aligned ds_load_b64 into the even-aligned VGPR pair WMMA consumes.
//
// Fragment layouts (ISA 7.12.2, f32):
//   A 16x4  : 2 VGPRs/lane. lanes 0-15: M=lane,  K = k+{0,1};
//                            lanes 16-31: M=lane-16, K = k+{2,3}
//   B 4x16  : 2 VGPRs/lane. lanes 0-15: N=lane,  K = k+{0,1};
//                            lanes 16-31: N=lane-16, K = k+{2,3}
//   C/D 16x16: 8 VGPRs. VGPR g: lanes 0-15 -> (M=g,   N=lane),
//                               lanes 16-31 -> (M=g+8, N=lane-16)
// ---------------------------------------------------------------------------
__global__ __launch_bounds__(256) void ptranse_wmma_gemm(const float* __restrict__ rw,
                                                         const float* __restrict__ rel_emb,
                                                         const int* __restrict__ r,
                                                         float* __restrict__ out) {
    extern __shared__ float bs[];                       // KCHUNK * DIM floats (interleaved)

    const int tid   = threadIdx.x;
    const int lane  = tid & 31;
    const int wave  = tid >> 5;                         // 0..7
    const int mtile = blockIdx.x * 8 + wave;            // 0..1023
    const int m0    = mtile * 16;

    const int lhalf = lane >> 4;                        // 0 | 1
    const int lmod  = lane & 15;                        // 0..15

    v8f c[8];
#pragma unroll
    for (int t = 0; t < 8; ++t) c[t] = (v8f)(0.0f);

    // This lane's A row (row-major [BATCH, NUM_REL]); K pair contiguous -> b64 load
    const float* arow = rw + (size_t)(m0 + lmod) * NUM_REL + 2 * lhalf;

    for (int kbase = 0; kbase < NUM_REL; kbase += KCHUNK) {
        // --- Cooperative stage with K-pair interleave ---
        {
            const int npairs = KCHUNK / 2;              // 150
            const int ntasks = npairs * (DIM / 4);      // 4800 tasks
            for (int task = tid; task < ntasks; task += 256) {
                const int p  = task >> 5;               // DIM/4 == 32 tasks per pair
                const int n4 = (task & 31) * 4;
                const float* r0p = rel_emb + (size_t)(kbase + 2 * p) * DIM + n4;
                const float4 r0 = *(const float4*)r0p;
                const float4 r1 = *(const float4*)(r0p + DIM);
                float2* dst = (float2*)(bs + (size_t)(p * DIM + n4) * 2);
                dst[0] = make_float2(r0.x, r1.x);
                dst[1] = make_float2(r0.y, r1.y);
                dst[2] = make_float2(r0.z, r1.z);
                dst[3] = make_float2(r0.w, r1.w);
            }
        }
        __syncthreads();

        // --- WMMA inner loop: A from global (b64), B from LDS (b64) ---
        for (int kk = 0; kk < KCHUNK; kk += 4) {
            v2f a = *(const v2f*)(arow + kbase + kk);   // A[M=lmod, k+2*lhalf+{0,1}]

            const int p = (kk >> 1) + lhalf;            // this lane-half's K pair
            const float* bb = bs + (size_t)(p * DIM + lmod) * 2;
#pragma unroll
            for (int t = 0; t < 8; ++t) {
                v2f b = *(const v2f*)(bb + t * 32);     // 16 cols * 2 floats per tile
                c[t] = __builtin_amdgcn_wmma_f32_16x16x4_f32(
                    /*neg_a=*/false, a, /*neg_b=*/false, b,
                    /*c_mod=*/(short)0, c[t], /*reuse_a=*/false, /*reuse_b=*/false);
            }
        }
        __syncthreads();
    }

    // Epilogue: out[row, col] = |P - rel_emb[r[row], col]|
#pragma unroll
    for (int g = 0; g < 8; ++g) {
        const int row = m0 + g + 8 * lhalf;
        const int rid = r[row];
        const float* rrow = rel_emb + (size_t)rid * DIM;
        float* orow = out + (size_t)row * DIM;
#pragma unroll
        for (int t = 0; t < 8; ++t) {
            const int col = t * 16 + lmod;
            orow[col] = fabsf(c[t][g] - rrow[col]);
        }
    }
}

// ---------------------------------------------------------------------------
// Launcher
// Inputs (setup_inputs order): r[i32,16384], path_batch[i32,1M],
//   path_rel[i32,1M], path_prob[f32,1M], rel_emb[f32,1200*128]
// Output: f32 [16384,128]. d_ws holds relation_weight [16384,1200] f32.
// ---------------------------------------------------------------------------
extern "C" void kernel_launch(void* const* d_in, const int* in_sizes, int n_in,
                              void* d_out, int out_size, void* d_ws, size_t ws_size,
                              hipStream_t stream) {
    const int*   r    = (const int*)d_in[0];
    const int*   pb   = (const int*)d_in[1];
    const int*   pr   = (const int*)d_in[2];
    const float* prob = (const float*)d_in[3];
    const float* rel  = (const float*)d_in[4];
    float*       out  = (float*)d_out;
    float*       rw   = (float*)d_ws;
    const int n_paths = in_sizes[1];

    const int n4 = (BATCH * NUM_REL) / 4;               // 4,915,200 float4
    ptranse_zero<<<2048, 256, 0, stream>>>((float4*)rw, n4);

    ptranse_scatter<<<(n_paths + 255) / 256, 256, 0, stream>>>(pb, pr, prob, rw, n_paths);

    const size_t lds_bytes = (size_t)KCHUNK * DIM * sizeof(float);   // 153600 B
    ptranse_wmma_gemm<<<BATCH / (16 * 8), 256, lds_bytes, stream>>>(rw, rel, r, out);
}